// MyActor_90494960926850
// MI455X (gfx1250) — compile-verified
//
#include <hip/hip_runtime.h>
#include <hip/hip_bf16.h>

// ---------------------------------------------------------------------------
// Problem: output(B=262144, N=16) = broadcast of a single 16-float vector,
// since the ODE state starts at zero and its dynamics are batch-independent
// (obs and t are unused by the reference math).
//
// Kernel 1: one wave integrates the 16-dim Kuramoto-style state for 10 steps.
//           The 16x16 coupling matvecs are done with V_WMMA_F32_16X16X4_F32.
// Kernel 2: bandwidth kernel; streams the 64-byte row pattern into 16 MB of
//           output with one float4 store per lane.
// ---------------------------------------------------------------------------

typedef __attribute__((ext_vector_type(2))) float v2f;
typedef __attribute__((ext_vector_type(8))) float v8f;

__global__ __launch_bounds__(32)
void kuramoto_state_kernel(const float* __restrict__ dt,
                           const float* __restrict__ mu,
                           const float* __restrict__ a,
                           const float* __restrict__ w,
                           const float* __restrict__ phi,
                           const float* __restrict__ freq,
                           float* __restrict__ out16)
{
    __shared__ float sP[16];   // p_j = r_j * sin(theta_j)
    __shared__ float sQ[16];   // q_j = r_j * cos(theta_j)

    const int lane    = threadIdx.x;        // 0..31, single wave
    const int i       = lane & 15;          // matrix row / state index
    const bool lowHalf = lane < 16;
    const int kbase   = lowHalf ? 0 : 2;    // K-pair owned by this lane half

    const float dts  = dt[0];
    const float mu_i = mu[i];
    const float a_i  = a[i];
    const float f_i  = freq[i];

    // ---- Constant B operands (WCᵀ and WSᵀ chunks), fixed across all steps.
    // B layout for f32 4x16 (2 VGPRs): lanes 0-15 hold K=0 (V0) / K=1 (V1),
    // lanes 16-31 hold K=2 (V0) / K=3 (V1), column N = lane%16.
    // B[k][n] = Mat[n][4*kb + k],  Mat = w .* cos(phi)  or  w .* sin(phi).
    float bWC0[4], bWC1[4], bWS0[4], bWS1[4];
    #pragma unroll
    for (int kb = 0; kb < 4; ++kb) {
        const int j0 = 4 * kb + kbase;
        const int j1 = j0 + 1;
        const float w0 = w[i * 16 + j0],  w1 = w[i * 16 + j1];
        const float p0 = phi[i * 16 + j0], p1 = phi[i * 16 + j1];
        bWC0[kb] = w0 * cosf(p0);
        bWC1[kb] = w1 * cosf(p1);
        bWS0[kb] = w0 * sinf(p0);
        bWS1[kb] = w1 * sinf(p1);
    }

    // State (owned by lanes 0-15); starts at zero.
    float r = 0.0f, rdot = 0.0f, theta = 0.0f;
    float s = 0.0f, c = 1.0f;

    #pragma unroll 1
    for (int step = 0; step < 10; ++step) {
        s = sinf(theta);
        c = cosf(theta);
        if (lowHalf) {
            sP[i] = r * s;
            sQ[i] = r * c;
        }
        __syncthreads();

        // ---- A operand: 16x4 f32, row M = lane%16, K-pair = kbase.
        // Rows 0-7 replicate p, rows 8-15 replicate q  =>  D rows M=0 / M=8
        // give (Mat.p)_n and (Mat.q)_n respectively in VGPR0.
        const float* sel = (i < 8) ? sP : sQ;

        v8f accWC = {0.f,0.f,0.f,0.f,0.f,0.f,0.f,0.f};
        v8f accWS = {0.f,0.f,0.f,0.f,0.f,0.f,0.f,0.f};
        #pragma unroll
        for (int kb = 0; kb < 4; ++kb) {
            v2f A;
            A.x = sel[4 * kb + kbase];
            A.y = sel[4 * kb + kbase + 1];
            v2f BWC; BWC.x = bWC0[kb]; BWC.y = bWC1[kb];
            v2f BWS; BWS.x = bWS0[kb]; BWS.y = bWS1[kb];
            accWC = __builtin_amdgcn_wmma_f32_16x16x4_f32(
                        false, A, false, BWC, (short)0, accWC, false, false);
            accWS = __builtin_amdgcn_wmma_f32_16x16x4_f32(
                        false, A, false, BWS, (short)0, accWS, false, false);
        }
        __syncthreads();   // done reading sP/sQ; safe to overwrite next step

        // D VGPR0: lanes 0-15 -> (Mat.p)_i ; lanes 16-31 -> (Mat.q)_i
        const float wcp = accWC[0];                 // lane<16: WC.p ; lane>=16: WC.q
        const float wsp = accWS[0];                 // lane<16: WS.p ; lane>=16: WS.q
        const float wcq = __shfl_xor(wcp, 16);      // lane<16 receives WC.q
        const float wsq = __shfl_xor(wsp, 16);      // lane<16 receives WS.q

        if (lowHalf) {
            const float coupling  = c * (wcp - wsq) - s * (wcq + wsp);
            const float theta_dot = f_i + coupling;
            const float rdd       = a_i * (a_i * 0.25f * (mu_i - r) - rdot);
            const float r_new     = r + dts * rdot;
            rdot  = rdot  + dts * rdd;
            theta = theta + dts * theta_dot;
            r     = r_new;
        }
    }

    if (lowHalf) {
        out16[i] = r + a_i * sinf(theta);
    }
}

// ---------------------------------------------------------------------------
// Bandwidth kernel: replicate the 64-byte row across the whole output.
// One float4 (global_store_b128) per lane; 4096 blocks x 256 threads covers
// 1,048,576 float4s = 16 MB exactly.
// ---------------------------------------------------------------------------
__global__ __launch_bounds__(256)
void broadcast_row_kernel(const float4* __restrict__ row4,
                          float4* __restrict__ out4,
                          int n4)
{
    const int idx = blockIdx.x * 256 + threadIdx.x;
    if (idx < n4) {
        out4[idx] = row4[idx & 3];   // row pattern repeats every 4 float4s
    }
}

extern "C" void kernel_launch(void* const* d_in, const int* in_sizes, int n_in,
                              void* d_out, int out_size, void* d_ws, size_t ws_size,
                              hipStream_t stream) {
    // setup_inputs order: obs, t, dt, mu, a, w, phi, frequencies
    const float* dt   = (const float*)d_in[2];
    const float* mu   = (const float*)d_in[3];
    const float* a    = (const float*)d_in[4];
    const float* w    = (const float*)d_in[5];
    const float* phi  = (const float*)d_in[6];
    const float* freq = (const float*)d_in[7];

    float* row = (float*)d_ws;          // 16 floats of scratch

    kuramoto_state_kernel<<<1, 32, 0, stream>>>(dt, mu, a, w, phi, freq, row);

    const int n4   = out_size / 4;      // number of float4 elements (1,048,576)
    const int grid = (n4 + 255) / 256;  // 4096
    broadcast_row_kernel<<<grid, 256, 0, stream>>>((const float4*)d_ws,
                                                   (float4*)d_out, n4);
}